// AttnDecoderRNN_49967649522276
// MI455X (gfx1250) — compile-verified
//
#include <hip/hip_runtime.h>
#include <hip/hip_bf16.h>
#include <math.h>

#define H 1024
#define V 50257
#define MAX_LEN 30

typedef __attribute__((ext_vector_type(2))) float v2f;
typedef __attribute__((ext_vector_type(8))) float v8f;

// ---------------------------------------------------------------------------
// Generic WMMA-based GEMV:  y[r] = act(dot(W[r,:], x) + bias[r])
// One wave (32 lanes) per 16-row tile. Uses V_WMMA_F32_16X16X4_F32 with the
// B matrix replicated across all 16 columns so every column of D equals W.x.
// A layout (32-bit, 16x4): lanes 0-15 -> M=0..15, K=k,k+1 ; lanes 16-31 ->
// M=0..15, K=k+2,k+3.  C/D layout: VGPR v, lanes 0-15 -> M=v, lanes 16-31 ->
// M=v+8, so lane 0 holds rows 0..7 and lane 16 holds rows 8..15.
// ---------------------------------------------------------------------------
__global__ void __launch_bounds__(32) wmma_gemv_kernel(
    const float* __restrict__ W, const float* __restrict__ x,
    const float* __restrict__ bias, float* __restrict__ y,
    int rows, int cols, int relu)
{
  const int lane = threadIdx.x & 31;
  const int row0 = blockIdx.x * 16;
  const bool lo  = lane < 16;

  int r = row0 + (lane & 15);
  if (r > rows - 1) r = rows - 1;          // clamp: garbage rows discarded at store
  const float* wrow = W + (size_t)r * cols + (lo ? 0 : 2);

  v8f acc = {};
  for (int k = 0; k < cols; k += 4) {
    v2f a;
    a.x = wrow[k];
    a.y = wrow[k + 1];
    float x0 = x[k + 0];
    float x1 = x[k + 1];
    float x2 = x[k + 2];
    float x3 = x[k + 3];
    v2f b;
    b.x = lo ? x0 : x2;
    b.y = lo ? x1 : x3;
    acc = __builtin_amdgcn_wmma_f32_16x16x4_f32(false, a, false, b,
                                                (short)0, acc, false, false);
  }

  if (lane == 0 || lane == 16) {
    const int base = row0 + (lo ? 0 : 8);
#pragma unroll
    for (int v = 0; v < 8; ++v) {
      int rr = base + v;
      if (rr < rows) {
        float val = acc[v] + bias[rr];
        if (relu) val = fmaxf(val, 0.0f);
        y[rr] = val;
      }
    }
  }
}

// ---------------------------------------------------------------------------
// Prep: embedding gather, attention scores + softmax, attn-applied, cat2.
// ---------------------------------------------------------------------------
__global__ void __launch_bounds__(256) attn_prep_kernel(
    const int* __restrict__ token, const float* __restrict__ hidden,
    const float* __restrict__ enc, const float* __restrict__ emb,
    const float* __restrict__ attn_W, const float* __restrict__ attn_b,
    float* __restrict__ cat2, float* __restrict__ out_attn)
{
  __shared__ float cat1[2 * H];
  __shared__ float scores[MAX_LEN];
  __shared__ float aw[MAX_LEN];
  const int tid  = threadIdx.x;
  const int lane = tid & 31;
  const int wave = tid >> 5;

  const long long tok = (long long)token[0];
  for (int i = tid; i < H; i += 256) {
    cat1[i]     = emb[(size_t)tok * H + i];
    cat1[H + i] = hidden[i];
  }
  __syncthreads();

  // scores[j] = attn_W[j,:] . cat1 + attn_b[j]   (one wave per row, strided)
  for (int j = wave; j < MAX_LEN; j += 8) {
    float s = 0.0f;
    for (int k = lane; k < 2 * H; k += 32)
      s = fmaf(attn_W[j * 2 * H + k], cat1[k], s);
    for (int off = 16; off > 0; off >>= 1) s += __shfl_down(s, off, 32);
    if (lane == 0) scores[j] = s + attn_b[j];
  }
  __syncthreads();

  // softmax over 30 positions (wave 0)
  if (wave == 0) {
    float v = (lane < MAX_LEN) ? scores[lane] : -1e30f;
    float mx = v;
    for (int off = 16; off > 0; off >>= 1) mx = fmaxf(mx, __shfl_xor(mx, off, 32));
    float e = (lane < MAX_LEN) ? expf(v - mx) : 0.0f;
    float sum = e;
    for (int off = 16; off > 0; off >>= 1) sum += __shfl_xor(sum, off, 32);
    float w = e / sum;
    if (lane < MAX_LEN) { aw[lane] = w; out_attn[lane] = w; }
  }
  __syncthreads();

  // attn_applied[h] = sum_j aw[j] * enc[j][h];  cat2 = [embedded, attn_applied]
  for (int h = tid; h < H; h += 256) {
    float acc = 0.0f;
    for (int j = 0; j < MAX_LEN; ++j)
      acc = fmaf(aw[j], enc[j * H + h], acc);
    cat2[h]     = cat1[h];
    cat2[H + h] = acc;
  }
}

// ---------------------------------------------------------------------------
// GRU gates (PyTorch order r,z,n) -> h_new
// ---------------------------------------------------------------------------
__global__ void __launch_bounds__(256) gru_gate_kernel(
    const float* __restrict__ gi, const float* __restrict__ gh,
    const float* __restrict__ hidden, float* __restrict__ h_new_out,
    float* __restrict__ h_new_ws)
{
  int i = blockIdx.x * blockDim.x + threadIdx.x;
  if (i < H) {
    float r = 1.0f / (1.0f + expf(-(gi[i] + gh[i])));
    float z = 1.0f / (1.0f + expf(-(gi[i + H] + gh[i + H])));
    float n = tanhf(gi[i + 2 * H] + r * gh[i + 2 * H]);
    float hv = (1.0f - z) * n + z * hidden[i];
    h_new_out[i] = hv;
    h_new_ws[i]  = hv;
  }
}

// ---------------------------------------------------------------------------
// log-softmax: logZ = max + log(sum(exp(l - max)))  (single block over V)
// ---------------------------------------------------------------------------
__global__ void __launch_bounds__(1024) logz_kernel(
    const float* __restrict__ logits, float* __restrict__ logz)
{
  __shared__ float red[1024];
  const int tid = threadIdx.x;

  float m = -1e30f;
  for (int v = tid; v < V; v += 1024) m = fmaxf(m, logits[v]);
  red[tid] = m;
  __syncthreads();
  for (int s = 512; s > 0; s >>= 1) {
    if (tid < s) red[tid] = fmaxf(red[tid], red[tid + s]);
    __syncthreads();
  }
  const float M = red[0];
  __syncthreads();

  float sum = 0.0f;
  for (int v = tid; v < V; v += 1024) sum += expf(logits[v] - M);
  red[tid] = sum;
  __syncthreads();
  for (int s = 512; s > 0; s >>= 1) {
    if (tid < s) red[tid] += red[tid + s];
    __syncthreads();
  }
  if (tid == 0) logz[0] = M + logf(red[0]);
}

__global__ void __launch_bounds__(256) logsub_kernel(
    float* __restrict__ out, const float* __restrict__ logz)
{
  int v = blockIdx.x * blockDim.x + threadIdx.x;
  if (v < V) out[v] -= logz[0];
}

// ---------------------------------------------------------------------------
extern "C" void kernel_launch(void* const* d_in, const int* in_sizes, int n_in,
                              void* d_out, int out_size, void* d_ws, size_t ws_size,
                              hipStream_t stream)
{
  (void)in_sizes; (void)n_in; (void)out_size; (void)ws_size;

  const int*   token  = (const int*)  d_in[0];
  const float* hidden = (const float*)d_in[1];
  const float* enc    = (const float*)d_in[2];
  const float* emb    = (const float*)d_in[3];
  const float* attn_W = (const float*)d_in[4];
  const float* attn_b = (const float*)d_in[5];
  const float* comb_W = (const float*)d_in[6];
  const float* comb_b = (const float*)d_in[7];
  const float* w_ih   = (const float*)d_in[8];
  const float* w_hh   = (const float*)d_in[9];
  const float* b_ih   = (const float*)d_in[10];
  const float* b_hh   = (const float*)d_in[11];
  const float* out_W  = (const float*)d_in[12];
  const float* out_b  = (const float*)d_in[13];

  float* out = (float*)d_out;   // [V output][H new_hidden][30 attn_weights]
  float* ws  = (float*)d_ws;
  float* cat2  = ws;            // 2048
  float* xrelu = ws + 2048;     // 1024
  float* gi    = ws + 3072;     // 3072
  float* gh    = ws + 6144;     // 3072
  float* hnew  = ws + 9216;     // 1024 (16B aligned copy for the big GEMV)
  float* logz  = ws + 10240;    // 1

  attn_prep_kernel<<<1, 256, 0, stream>>>(token, hidden, enc, emb,
                                          attn_W, attn_b, cat2, out + V + H);
  wmma_gemv_kernel<<<(H + 15) / 16, 32, 0, stream>>>(comb_W, cat2, comb_b,
                                                     xrelu, H, 2 * H, 1);
  wmma_gemv_kernel<<<(3 * H + 15) / 16, 32, 0, stream>>>(w_ih, xrelu, b_ih,
                                                         gi, 3 * H, H, 0);
  wmma_gemv_kernel<<<(3 * H + 15) / 16, 32, 0, stream>>>(w_hh, hidden, b_hh,
                                                         gh, 3 * H, H, 0);
  gru_gate_kernel<<<(H + 255) / 256, 256, 0, stream>>>(gi, gh, hidden,
                                                       out + V, hnew);
  wmma_gemv_kernel<<<(V + 15) / 16, 32, 0, stream>>>(out_W, hnew, out_b,
                                                     out, V, H, 0);
  logz_kernel<<<1, 1024, 0, stream>>>(out, logz);
  logsub_kernel<<<(V + 255) / 256, 256, 0, stream>>>(out, logz);
}